// Finufft1D1_82094004896528
// MI455X (gfx1250) — compile-verified
//
#include <hip/hip_runtime.h>
#include <math.h>

typedef __attribute__((ext_vector_type(2))) float v2f;
typedef __attribute__((ext_vector_type(8))) float v8f;

#define NWG 128
#define PTS_PER_WG 8192
#define KB 32
#define BLOCKS_PER_WG (PTS_PER_WG / KB)
#define NMODES 4096
#define PITCH 34          // floats per half-row (KB + 2): breaks bank conflicts
#define ROW (2 * PITCH)   // [ re x34 | im x34 ] per matrix row

static __device__ inline void cmul(float& ar, float& ai, float br, float bi) {
    float r = ar * br - ai * bi;
    float i = ar * bi + ai * br;
    ar = r; ai = i;
}

static __device__ inline v8f wmma_f32(v2f a, v2f b, v8f c) {
    // D(16x16,f32) = A(16x4,f32) x B(4x16,f32) + C
    return __builtin_amdgcn_wmma_f32_16x16x4_f32(
        /*neg_a=*/false, a, /*neg_b=*/false, b,
        /*c_mod=*/(short)0, c, /*reuse_a=*/false, /*reuse_b=*/false);
}

// f[k] = sum_j c_j exp(i k x_j), k = m - 2048, m = m0 + 64*m1:
//   A[m1][j] = exp(i*64*(m1-32)*x_j)   (absorbs the -2048 shift)
//   B[j][m0] = c_j * exp(i*m0*x_j)
//   F[m1][m0] = sum_j A*B  -> complex GEMM [64 x M] x [M x 64] via f32 WMMA.
__global__ __launch_bounds__(256) void nufft_partial_kernel(
    const float* __restrict__ points,
    const float* __restrict__ vre,
    const float* __restrict__ vim,
    float* __restrict__ partial,   // [NWG][NMODES][2]
    int M)
{
    __shared__ __align__(16) float As[64][ROW];  // [m1][re|im][k]
    __shared__ __align__(16) float Bs[64][ROW];  // [m0][re|im][k]

    const int tid  = threadIdx.x;
    const int lane = tid & 31;
    const int wv   = tid >> 5;

    // Producer role: point-in-block pb, power group g (g uniform per wave)
    const int pb = tid & (KB - 1);
    const int g  = tid >> 5;                   // owns m-range [8g, 8g+8)
    const float bcoef = (float)(8 * g);        // B base multiplier
    const float acoef = (float)(512 * g - 2048); // A base multiplier

    // Consumer role: wave wv owns output tiles (t1, t0a) and (t1, t0a+1)
    const int t1    = wv >> 1;
    const int t0a   = (wv & 1) * 2;
    const int mrow  = lane & 15;
    const int khalf = (lane >> 4) * 2;

    v8f acc_rr[2], acc_ii[2], acc_ri[2], acc_ir[2];
    const v8f zero = {0.f, 0.f, 0.f, 0.f, 0.f, 0.f, 0.f, 0.f};
    for (int t = 0; t < 2; ++t) {
        acc_rr[t] = zero; acc_ii[t] = zero; acc_ri[t] = zero; acc_ir[t] = zero;
    }

    const long base0 = (long)blockIdx.x * PTS_PER_WG;

    for (int blk = 0; blk < BLOCKS_PER_WG; ++blk) {
        // ---------------- produce phase factors into LDS ----------------
        long j = base0 + (long)blk * KB + pb;
        float x = 0.f, cr = 0.f, ci = 0.f;
        if (j < M) { x = points[j]; cr = vre[j]; ci = vim[j]; }

        float s1, c1, s64, c64, sb, cb, sa, ca;
        sincosf(x,          &s1,  &c1);   // step for B ladder
        sincosf(64.0f * x,  &s64, &c64);  // step for A ladder
        sincosf(bcoef * x,  &sb,  &cb);   // e^{i*8g*x}
        sincosf(acoef * x,  &sa,  &ca);   // e^{i*(512g-2048)*x}

        // B ladder: w = c * e^{i*8g*x}, then step by e^{i*x}
        {
            float wr = cr * cb - ci * sb;
            float wi = cr * sb + ci * cb;
            #pragma unroll
            for (int k2 = 0; k2 < 8; ++k2) {
                int m = g * 8 + k2;
                Bs[m][pb]         = wr;
                Bs[m][PITCH + pb] = wi;
                cmul(wr, wi, c1, s1);
            }
        }
        // A ladder: p = e^{i*(512g-2048)*x}, then step by e^{i*64x}
        {
            float pr = ca, pi = sa;
            #pragma unroll
            for (int k2 = 0; k2 < 8; ++k2) {
                int m = g * 8 + k2;
                As[m][pb]         = pr;
                As[m][PITCH + pb] = pi;
                cmul(pr, pi, c64, s64);
            }
        }
        __syncthreads();

        // ---------------- consume: WMMA over K = KB points ----------------
        #pragma unroll
        for (int kk = 0; kk < KB / 4; ++kk) {
            int ke = kk * 4 + khalf;  // even -> 8B aligned b64 loads
            const float* arow = &As[t1 * 16 + mrow][0];
            v2f ar = *(const v2f*)&arow[ke];           // A re, K = ke, ke+1
            v2f ai = *(const v2f*)&arow[PITCH + ke];   // A im
            #pragma unroll
            for (int tt = 0; tt < 2; ++tt) {
                const float* brow = &Bs[(t0a + tt) * 16 + mrow][0];
                v2f br = *(const v2f*)&brow[ke];
                v2f bi = *(const v2f*)&brow[PITCH + ke];
                acc_rr[tt] = wmma_f32(ar, br, acc_rr[tt]);
                acc_ii[tt] = wmma_f32(ai, bi, acc_ii[tt]);
                acc_ri[tt] = wmma_f32(ar, bi, acc_ri[tt]);
                acc_ir[tt] = wmma_f32(ai, br, acc_ir[tt]);
            }
        }
        __syncthreads();
    }

    // ---------------- write this WG's partial 64x64 complex result ----------------
    // C/D layout: VGPR v holds M = v + 8*(lane>=16), N = lane%16
    float* p = partial + (size_t)blockIdx.x * (NMODES * 2);
    #pragma unroll
    for (int tt = 0; tt < 2; ++tt) {
        #pragma unroll
        for (int v = 0; v < 8; ++v) {
            int Mr = v + 8 * (lane >> 4);
            int Nc = lane & 15;
            int m = (t1 * 16 + Mr) * 64 + (t0a + tt) * 16 + Nc;
            p[2 * m + 0] = acc_rr[tt][v] - acc_ii[tt][v];
            p[2 * m + 1] = acc_ri[tt][v] + acc_ir[tt][v];
        }
    }
}

// Kernel 2: deterministic reduction of the NWG partials + FFT-order permutation.
__global__ __launch_bounds__(256) void nufft_reduce_kernel(
    const float* __restrict__ partial, float* __restrict__ out)
{
    int m = blockIdx.x * blockDim.x + threadIdx.x;
    if (m >= NMODES) return;
    float re = 0.f, im = 0.f;
    for (int w = 0; w < NWG; ++w) {
        const float* p = partial + (size_t)w * (NMODES * 2) + 2 * m;
        re += p[0];
        im += p[1];
    }
    int n = m ^ 2048;   // out index for mode k = m - 2048
    out[2 * n + 0] = re;
    out[2 * n + 1] = im;
}

extern "C" void kernel_launch(void* const* d_in, const int* in_sizes, int n_in,
                              void* d_out, int out_size, void* d_ws, size_t ws_size,
                              hipStream_t stream) {
    const float* points = (const float*)d_in[0];
    const float* vre    = (const float*)d_in[1];
    const float* vim    = (const float*)d_in[2];
    int M = in_sizes[0];

    float* partial = (float*)d_ws;   // NWG * 4096 * 2 floats = 4 MB

    nufft_partial_kernel<<<NWG, 256, 0, stream>>>(points, vre, vim, partial, M);
    nufft_reduce_kernel<<<(NMODES + 255) / 256, 256, 0, stream>>>(partial, (float*)d_out);
}